// NormLearningEngine_53446573031658
// MI455X (gfx1250) — compile-verified
//
#include <hip/hip_runtime.h>
#include <math.h>

typedef float v2f __attribute__((ext_vector_type(2)));
typedef float v4f __attribute__((ext_vector_type(4)));
typedef float v8f __attribute__((ext_vector_type(8)));

#define DD    1024
#define HH    256
#define KK_   64
#define CTXW  16
#define BB    64
#define TT    2048
#define TSPLIT 32
#define ROWS_PER_SPLIT (TT / TSPLIT)   /* 64 */

/* workspace layout (float offsets) */
#define WS_PARTIAL  0
#define WS_ACTION   (BB * TSPLIT * DD)          /* 2,097,152 */
#define WS_CTXENC   (WS_ACTION + BB * DD)       /* + 65,536  */
#define WS_NW       (WS_CTXENC + DD)
#define WS_CVEC     (WS_NW + KK_)
#define WS_SVCTX    (WS_CVEC + 512)
#define WS_AW       (WS_SVCTX + HH)
#define WS_PROTO    (WS_AW + BB * 512)

__device__ __forceinline__ float gelu_f(float v) {
    return 0.5f * v * (1.0f + erff(v * 0.70710678118654752440f));
}
__device__ __forceinline__ float sigmoid_f(float v) {
    return 1.0f / (1.0f + expf(-v));
}

/* ---- Phase 1a: partial sums of x over t (streaming, b128 NT loads) ----
 * x is 512 MB read exactly once (> 192 MB L2): non-temporal hint keeps the
 * stream from evicting the small weight matrices the later phases re-read.
 */
__global__ void k_partial_mean(const float* __restrict__ x, float* __restrict__ ws) {
    const int b  = blockIdx.x;
    const int sp = blockIdx.y;
    const int t  = threadIdx.x;
    const v4f* xp = (const v4f*)(x + (size_t)b * TT * DD
                                   + (size_t)sp * ROWS_PER_SPLIT * DD) + t;
    v4f acc = {0.f, 0.f, 0.f, 0.f};
#pragma unroll 4
    for (int r = 0; r < ROWS_PER_SPLIT; ++r) {
        v4f v = __builtin_nontemporal_load(xp + (size_t)r * (DD / 4));
        acc += v;
    }
    v4f* o = (v4f*)(ws + WS_PARTIAL + (size_t)(b * TSPLIT + sp) * DD) + t;
    *o = acc;
}

/* ---- Phase 1b: reduce partials -> action = x.mean(axis=1) ---- */
__global__ void k_reduce_mean(float* __restrict__ ws) {
    const int b = blockIdx.x;
    const int t = threadIdx.x;
    const v4f* p = (const v4f*)(ws + WS_PARTIAL + (size_t)b * TSPLIT * DD) + t;
    v4f acc = {0.f, 0.f, 0.f, 0.f};
    for (int s = 0; s < TSPLIT; ++s) {
        acc += p[(size_t)s * (DD / 4)];
    }
    acc *= (1.0f / (float)TT);
    ((v4f*)(ws + WS_ACTION + (size_t)b * DD))[t] = acc;
}

/* ---- Phase 2: whole context pipeline in one workgroup (batch-invariant) ---- */
__global__ void k_context(const float* __restrict__ cb,
                          const float* __restrict__ ce_w1, const float* __restrict__ ce_b1,
                          const float* __restrict__ ce_w2, const float* __restrict__ ce_b2,
                          const float* __restrict__ rms_w,
                          const float* __restrict__ nm_w1, const float* __restrict__ nm_b1,
                          const float* __restrict__ ns_w1, const float* __restrict__ ns_b1,
                          const float* __restrict__ ns_w2, const float* __restrict__ ns_b2,
                          const float* __restrict__ sv_w1, const float* __restrict__ sv_b1,
                          float* __restrict__ ws) {
    const int t = threadIdx.x;
    __shared__ float s_ctx[DD];
    __shared__ float s_h[HH];
    __shared__ float s_enc[DD];
    __shared__ float s_red[256];
    __shared__ float s_log[KK_];

    /* ctx = mean over CTXW rows */
    for (int i = 0; i < 4; ++i) {
        int d = t + 256 * i;
        float s = 0.f;
        for (int c = 0; c < CTXW; ++c) s += cb[c * DD + d];
        s_ctx[d] = s * (1.0f / (float)CTXW);
    }
    __syncthreads();

    /* h1 = gelu(ctx @ ce_w1 + ce_b1) */
    {
        float acc = ce_b1[t];
        for (int d = 0; d < DD; ++d) acc += s_ctx[d] * ce_w1[(size_t)d * HH + t];
        s_h[t] = gelu_f(acc);
    }
    __syncthreads();

    /* ctx_pre = h1 @ ce_w2 + ce_b2 ; RMS norm */
    float pre[4];
    float ssq = 0.f;
    for (int i = 0; i < 4; ++i) {
        int d = t + 256 * i;
        float acc = ce_b2[d];
        for (int h = 0; h < HH; ++h) acc += s_h[h] * ce_w2[(size_t)h * DD + d];
        pre[i] = acc;
        ssq += acc * acc;
    }
    s_red[t] = ssq;
    __syncthreads();
    for (int s = 128; s > 0; s >>= 1) {
        if (t < s) s_red[t] += s_red[t + s];
        __syncthreads();
    }
    const float rinv = rsqrtf(s_red[0] / (float)DD + 1e-6f);
    for (int i = 0; i < 4; ++i) {
        int d = t + 256 * i;
        float e = pre[i] * rinv * rms_w[d];
        s_enc[d] = e;
        ws[WS_CTXENC + d] = e;
    }
    __syncthreads();

    /* norm_weights = softmax(gelu(ctx_enc @ ns_w1 + b) @ ns_w2 + b) */
    {
        float acc = ns_b1[t];
        for (int d = 0; d < DD; ++d) acc += s_enc[d] * ns_w1[(size_t)d * HH + t];
        s_h[t] = gelu_f(acc);
    }
    __syncthreads();
    if (t < KK_) {
        float acc = ns_b2[t];
        for (int h = 0; h < HH; ++h) acc += s_h[h] * ns_w2[(size_t)h * KK_ + t];
        s_log[t] = acc;
    }
    __syncthreads();
    if (t < KK_) {
        float mx = -1e30f;
        for (int k = 0; k < KK_; ++k) mx = fmaxf(mx, s_log[k]);
        float sum = 0.f;
        for (int k = 0; k < KK_; ++k) sum += expf(s_log[k] - mx);
        ws[WS_NW + t] = expf(s_log[t] - mx) / sum;
    }

    /* cvec = ctx_enc @ wc + nm_b1  (wc = nm_w1 rows [0,1024)) */
    for (int i = 0; i < 2; ++i) {
        int j = t + 256 * i;
        float acc = nm_b1[j];
        for (int d = 0; d < DD; ++d) acc += s_enc[d] * nm_w1[(size_t)d * 512 + j];
        ws[WS_CVEC + j] = acc;
    }

    /* sv_ctx = ctx_enc @ sv_w1[:1024] + sv_b1 */
    {
        float acc = sv_b1[t];
        for (int d = 0; d < DD; ++d) acc += s_enc[d] * sv_w1[(size_t)d * HH + t];
        ws[WS_SVCTX + t] = acc;
    }
}

/* ---- Phase 3: fp32 WMMA GEMMs: aw = action@wa, proto = prototypes@wp ----
 * 16x16 output tile per wave, K-loop step 4 with V_WMMA_F32_16X16X4_F32.
 * A layout (ISA 16x4 f32): lanes 0-15 -> K0/K1, lanes 16-31 -> K2/K3, M = lane%16.
 * B mirrored with N = lane%16.
 */
__global__ void k_wmma_gemms(const float* __restrict__ protos,
                             const float* __restrict__ nm_w1,
                             float* __restrict__ ws) {
    const int wave = (blockIdx.x * blockDim.x + threadIdx.x) >> 5;   /* 0..255 */
    const int lane = threadIdx.x & 31;
    const int which = wave >> 7;          /* 0: action@wa, 1: protos@wp */
    const int tile  = wave & 127;
    const int mt = tile >> 5;             /* 0..3  (M/16)   */
    const int nt = tile & 31;             /* 0..31 (512/16) */

    const float* A = which ? protos : (ws + WS_ACTION);
    const float* W = nm_w1 + (size_t)(which ? 2048 : 1024) * 512;
    float* Out = ws + (which ? WS_PROTO : WS_AW);

    const int mrow = mt * 16 + (lane & 15);
    const int ncol = nt * 16 + (lane & 15);
    const int kk = (lane >> 4) * 2;       /* 0 or 2 */
    const float* Arow = A + (size_t)mrow * DD;

    v8f acc = {0.f, 0.f, 0.f, 0.f, 0.f, 0.f, 0.f, 0.f};
    for (int k = 0; k < DD; k += 4) {
        v2f a = *(const v2f*)(Arow + k + kk);
        v2f b;
        b.x = W[(size_t)(k + kk) * 512 + ncol];
        b.y = W[(size_t)(k + kk + 1) * 512 + ncol];
        acc = __builtin_amdgcn_wmma_f32_16x16x4_f32(
            false, a, false, b, (short)0, acc, false, false);
    }
    /* C/D layout: lane<16 -> M = r, N = lane ; lane>=16 -> M = 8+r, N = lane-16 */
    const int mbase = mt * 16 + ((lane >> 4) ? 8 : 0);
    const int col = nt * 16 + (lane & 15);
#pragma unroll
    for (int r = 0; r < 8; ++r)
        Out[(size_t)(mbase + r) * 512 + col] = acc[r];
}

/* ---- Phase 4: per-batch epilogue ---- */
__global__ void k_final(const float* __restrict__ nm_w2, const float* __restrict__ nm_b2,
                        const float* __restrict__ sv_w1, const float* __restrict__ sv_w2,
                        const float* __restrict__ sv_b2,
                        const float* __restrict__ ws, float* __restrict__ out) {
    const int b = blockIdx.x;
    const int t = threadIdx.x;
    __shared__ float s_pre[512];
    __shared__ float s_red[256];
    __shared__ float s_w[KK_];

    const float* aw   = ws + WS_AW + (size_t)b * 512;
    const float* cvec = ws + WS_CVEC;
    s_pre[t]       = aw[t]       + cvec[t];
    s_pre[t + 256] = aw[t + 256] + cvec[t + 256];
    __syncthreads();

    /* conformance: 4 threads per k, each sums 128 of the 512 hidden */
    {
        const int k = t >> 2, part = t & 3;
        const float* pr = ws + WS_PROTO + (size_t)k * 512;
        float sum = 0.f;
        const int j0 = part * 128;
        for (int j = j0; j < j0 + 128; ++j) {
            float v = s_pre[j] + pr[j];
            sum += gelu_f(v) * nm_w2[j];
        }
        s_red[t] = sum;
    }
    __syncthreads();
    if (t < KK_) {
        float dot = s_red[4 * t] + s_red[4 * t + 1] + s_red[4 * t + 2] + s_red[4 * t + 3]
                    + nm_b2[0];
        s_w[t] = sigmoid_f(dot) * ws[WS_NW + t];
    }
    __syncthreads();
    float wc_local = 0.f;
    if (t == 0) {
        for (int k = 0; k < KK_; ++k) wc_local += s_w[k];
    }

    /* severity: gelu(sv_ctx + action@sv_w1[1024:]) @ sv_w2 + b -> sigmoid */
    const float* act = ws + WS_ACTION + (size_t)b * DD;
    float acc = ws[WS_SVCTX + t];
    for (int d = 0; d < DD; ++d)
        acc += act[d] * sv_w1[(size_t)(1024 + d) * HH + t];
    __syncthreads();               /* s_red reads above complete before overwrite */
    s_red[t] = gelu_f(acc) * sv_w2[t];
    __syncthreads();
    for (int s = 128; s > 0; s >>= 1) {
        if (t < s) s_red[t] += s_red[t + s];
        __syncthreads();
    }
    if (t == 0) {
        const float sev  = sigmoid_f(s_red[0] + sv_b2[0]);
        const float wc   = wc_local;
        const float viol = 1.0f - wc;
        out[b]            = viol * sev * 0.1f;  /* norm_penalty  */
        out[64 + b]       = wc;                 /* weighted_conf */
        out[128 + b]      = viol;               /* violation     */
        out[192 + b]      = sev;                /* severity      */
    }
}

extern "C" void kernel_launch(void* const* d_in, const int* in_sizes, int n_in,
                              void* d_out, int out_size, void* d_ws, size_t ws_size,
                              hipStream_t stream) {
    (void)in_sizes; (void)n_in; (void)out_size; (void)ws_size;
    const float* x      = (const float*)d_in[0];
    const float* cb     = (const float*)d_in[1];
    const float* protos = (const float*)d_in[2];
    const float* ce_w1  = (const float*)d_in[3];
    const float* ce_b1  = (const float*)d_in[4];
    const float* ce_w2  = (const float*)d_in[5];
    const float* ce_b2  = (const float*)d_in[6];
    const float* rms_w  = (const float*)d_in[7];
    const float* nm_w1  = (const float*)d_in[8];
    const float* nm_b1  = (const float*)d_in[9];
    const float* nm_w2  = (const float*)d_in[10];
    const float* nm_b2  = (const float*)d_in[11];
    const float* ns_w1  = (const float*)d_in[12];
    const float* ns_b1  = (const float*)d_in[13];
    const float* ns_w2  = (const float*)d_in[14];
    const float* ns_b2  = (const float*)d_in[15];
    const float* sv_w1  = (const float*)d_in[16];
    const float* sv_b1  = (const float*)d_in[17];
    const float* sv_w2  = (const float*)d_in[18];
    const float* sv_b2  = (const float*)d_in[19];
    float* ws  = (float*)d_ws;
    float* out = (float*)d_out;

    k_partial_mean<<<dim3(BB, TSPLIT), 256, 0, stream>>>(x, ws);
    k_reduce_mean<<<dim3(BB), 256, 0, stream>>>(ws);
    k_context<<<1, 256, 0, stream>>>(cb, ce_w1, ce_b1, ce_w2, ce_b2, rms_w,
                                     nm_w1, nm_b1, ns_w1, ns_b1, ns_w2, ns_b2,
                                     sv_w1, sv_b1, ws);
    k_wmma_gemms<<<32, 256, 0, stream>>>(protos, nm_w1, ws);
    k_final<<<BB, 256, 0, stream>>>(nm_w2, nm_b2, sv_w1, sv_w2, sv_b2, ws, out);
}